// InterMetaPathAggregator_39410619908634
// MI455X (gfx1250) — compile-verified
//
#include <hip/hip_runtime.h>
#include <hip/hip_bf16.h>

#define N_NODES 100000
#define D_HID   64
#define H_HEADS 4
#define HD      256      // H*D
#define E_EDGES 400000
#define L_PATH  3
#define NT_TGT  50000

typedef float v2f __attribute__((ext_vector_type(2)));
typedef float v8f __attribute__((ext_vector_type(8)));

// ---------------------------------------------------------------------------
// WMMA f32 GEMM: C[M,Nc] = A[M,K] @ B[K,Nc], row-major, M%16==0, Nc%16==0, K%4==0
// One wave per 16x16 output tile; V_WMMA_F32_16X16X4_F32 along K.
// A 16x4 layout: lanes 0-15 -> M=lane, K={0,1}; lanes 16-31 -> M=lane-16, K={2,3}.
// ---------------------------------------------------------------------------
__global__ void k_wmma_gemm(const float* __restrict__ A, const float* __restrict__ B,
                            float* __restrict__ C, int M, int K, int Nc) {
  int wave = blockIdx.x * (blockDim.x >> 5) + (threadIdx.x >> 5);
  int tilesN = Nc >> 4;
  int tiles  = (M >> 4) * tilesN;
  if (wave >= tiles) return;                 // wave-uniform: EXEC stays all-ones
  int tm = wave / tilesN, tn = wave % tilesN;
  int lane = threadIdx.x & 31, half = lane >> 4, l16 = lane & 15;
  int row = (tm << 4) + l16;
  int col = (tn << 4) + l16;
  v8f c = {0.f,0.f,0.f,0.f,0.f,0.f,0.f,0.f};
  for (int k = 0; k < K; k += 4) {
    int ka = k + (half << 1);
    v2f a, b;
    a.x = A[(size_t)row * K + ka];
    a.y = A[(size_t)row * K + ka + 1];
    b.x = B[(size_t)ka * Nc + col];
    b.y = B[(size_t)(ka + 1) * Nc + col];
    c = __builtin_amdgcn_wmma_f32_16x16x4_f32(false, a, false, b, (short)0, c, false, false);
  }
  int rbase = (tm << 4) + (half << 3);
  for (int r = 0; r < 8; ++r)
    C[(size_t)(rbase + r) * Nc + col] = c[r];
}

// ---------------------------------------------------------------------------
// tanh(h_p @ Ws + bs) column-sum reduction (fused row-mean numerator).
// A: [NT,256], Ws: [256,64] -> accumulates sum over rows of tanh() into spsum[64].
// ---------------------------------------------------------------------------
__global__ void k_wmma_tanh_reduce(const float* __restrict__ A, const float* __restrict__ Ws,
                                   const float* __restrict__ bs, float* __restrict__ spsum) {
  const int K = HD, Nc = D_HID;
  int wave = blockIdx.x * (blockDim.x >> 5) + (threadIdx.x >> 5);
  int tilesN = Nc >> 4;
  int tiles  = (NT_TGT >> 4) * tilesN;
  if (wave >= tiles) return;
  int tm = wave / tilesN, tn = wave % tilesN;
  int lane = threadIdx.x & 31, half = lane >> 4, l16 = lane & 15;
  int row = (tm << 4) + l16;
  int col = (tn << 4) + l16;
  v8f c = {0.f,0.f,0.f,0.f,0.f,0.f,0.f,0.f};
  for (int k = 0; k < K; k += 4) {
    int ka = k + (half << 1);
    v2f a, b;
    a.x = A[(size_t)row * K + ka];
    a.y = A[(size_t)row * K + ka + 1];
    b.x = Ws[(size_t)ka * Nc + col];
    b.y = Ws[(size_t)(ka + 1) * Nc + col];
    c = __builtin_amdgcn_wmma_f32_16x16x4_f32(false, a, false, b, (short)0, c, false, false);
  }
  float bias = bs[col];
  float local = 0.f;
  for (int r = 0; r < 8; ++r) local += tanhf(c[r] + bias);
  local += __shfl_xor(local, 16, 32);        // lanes L and L+16 hold same column
  if (half == 0) atomicAdd(&spsum[col], local);
}

// ---------------------------------------------------------------------------
// Final: relu((beta0*h0 + beta1*h1) @ Wo + bo) -> out [NT,64]
// A fragments built on the fly from the beta-weighted combine.
// ---------------------------------------------------------------------------
__global__ void k_wmma_final(const float* __restrict__ h0, const float* __restrict__ h1,
                             const float* __restrict__ beta, const float* __restrict__ Wo,
                             const float* __restrict__ bo, float* __restrict__ out) {
  const int K = HD, Nc = D_HID;
  int wave = blockIdx.x * (blockDim.x >> 5) + (threadIdx.x >> 5);
  int tilesN = Nc >> 4;
  int tiles  = (NT_TGT >> 4) * tilesN;
  if (wave >= tiles) return;
  int tm = wave / tilesN, tn = wave % tilesN;
  int lane = threadIdx.x & 31, half = lane >> 4, l16 = lane & 15;
  int row = (tm << 4) + l16;
  int col = (tn << 4) + l16;
  float b0 = beta[0], b1 = beta[1];
  v8f c = {0.f,0.f,0.f,0.f,0.f,0.f,0.f,0.f};
  for (int k = 0; k < K; k += 4) {
    int ka = k + (half << 1);
    v2f a, b;
    a.x = b0 * h0[(size_t)row * K + ka]     + b1 * h1[(size_t)row * K + ka];
    a.y = b0 * h0[(size_t)row * K + ka + 1] + b1 * h1[(size_t)row * K + ka + 1];
    b.x = Wo[(size_t)ka * Nc + col];
    b.y = Wo[(size_t)(ka + 1) * Nc + col];
    c = __builtin_amdgcn_wmma_f32_16x16x4_f32(false, a, false, b, (short)0, c, false, false);
  }
  float bias = bo[col];
  int rbase = (tm << 4) + (half << 3);
  for (int r = 0; r < 8; ++r)
    out[(size_t)(rbase + r) * Nc + col] = fmaxf(c[r] + bias, 0.f);
}

// ---------------------------------------------------------------------------
// Per-node score factorization: s_src[n,h] = sum_d a_src[h,d]*xp[n,h,d] (same for dst).
// One wave per node, 2 elements per lane, wave shuffle reduction.
// ---------------------------------------------------------------------------
__global__ void k_node_scores(const float* __restrict__ xp, const float* __restrict__ a_src,
                              const float* __restrict__ a_dst,
                              float* __restrict__ ssrc, float* __restrict__ sdst) {
  int n = blockIdx.x * (blockDim.x >> 5) + (threadIdx.x >> 5);
  if (n >= N_NODES) return;
  int lane = threadIdx.x & 31;
  for (int h = 0; h < H_HEADS; ++h) {
    float x1 = xp[(size_t)n * HD + h * 64 + lane];
    float x2 = xp[(size_t)n * HD + h * 64 + lane + 32];
    float ps = a_src[h * 64 + lane] * x1 + a_src[h * 64 + lane + 32] * x2;
    float pd = a_dst[h * 64 + lane] * x1 + a_dst[h * 64 + lane + 32] * x2;
    for (int off = 16; off > 0; off >>= 1) {
      ps += __shfl_xor(ps, off, 32);
      pd += __shfl_xor(pd, off, 32);
    }
    if (lane == 0) { ssrc[n * H_HEADS + h] = ps; sdst[n * H_HEADS + h] = pd; }
  }
}

// ---------------------------------------------------------------------------
// Intra-metapath mean + per-edge a_edge dot term. One wave per edge.
// ---------------------------------------------------------------------------
__global__ void k_agg_edge(const float* __restrict__ x, const int* __restrict__ mp,
                           const float* __restrict__ a_edge,
                           float* __restrict__ agg, float* __restrict__ esc) {
  int e = blockIdx.x * (blockDim.x >> 5) + (threadIdx.x >> 5);
  if (e >= E_EDGES) return;
  int lane = threadIdx.x & 31;
  int i0 = mp[e * 3 + 0], i1 = mp[e * 3 + 1], i2 = mp[e * 3 + 2];
  const float inv3 = 1.0f / 3.0f;
  float v1 = (x[(size_t)i0 * 64 + lane]      + x[(size_t)i1 * 64 + lane]      + x[(size_t)i2 * 64 + lane])      * inv3;
  float v2 = (x[(size_t)i0 * 64 + lane + 32] + x[(size_t)i1 * 64 + lane + 32] + x[(size_t)i2 * 64 + lane + 32]) * inv3;
  agg[(size_t)e * 64 + lane]      = v1;
  agg[(size_t)e * 64 + lane + 32] = v2;
  for (int h = 0; h < H_HEADS; ++h) {
    float p = a_edge[h * 64 + lane] * v1 + a_edge[h * 64 + lane + 32] * v2;
    for (int off = 16; off > 0; off >>= 1) p += __shfl_xor(p, off, 32);
    if (lane == 0) esc[(size_t)e * H_HEADS + h] = p;
  }
}

// ---------------------------------------------------------------------------
// e = leaky_relu(edge_term + s_src[src] + s_dst[dst]); segment max into mmax[dst].
// Float atomic max via sign-split int/uint atomics (init = -inf).
// ---------------------------------------------------------------------------
__global__ void k_edge_score_max(float* __restrict__ esc, const int* __restrict__ src,
                                 const int* __restrict__ dst, const float* __restrict__ ssrc,
                                 const float* __restrict__ sdst, float* __restrict__ mmax) {
  int idx = blockIdx.x * blockDim.x + threadIdx.x;
  if (idx >= E_EDGES * H_HEADS) return;
  int e = idx >> 2, h = idx & 3;
  float v = esc[idx] + ssrc[src[e] * H_HEADS + h] + sdst[dst[e] * H_HEADS + h];
  v = v > 0.f ? v : 0.2f * v;
  esc[idx] = v;
  float* addr = &mmax[dst[e] * H_HEADS + h];
  if (v >= 0.f) atomicMax((int*)addr, __float_as_int(v));
  else          atomicMin((unsigned int*)addr, __float_as_uint(v));
}

__global__ void k_edge_exp(float* __restrict__ esc, const int* __restrict__ dst,
                           const float* __restrict__ mmax, float* __restrict__ den) {
  int idx = blockIdx.x * blockDim.x + threadIdx.x;
  if (idx >= E_EDGES * H_HEADS) return;
  int e = idx >> 2, h = idx & 3;
  float w = __expf(esc[idx] - mmax[dst[e] * H_HEADS + h]);
  esc[idx] = w;
  atomicAdd(&den[dst[e] * H_HEADS + h], w);
}

// ---------------------------------------------------------------------------
// Message scatter: out[dst,h,d] += alpha * (xp[src,h,d] + agg[e,d]).
// One 256-thread block per edge; non-returning global_atomic_add_f32.
// ---------------------------------------------------------------------------
__global__ void k_scatter(const float* __restrict__ esc, const float* __restrict__ den,
                          const int* __restrict__ src, const int* __restrict__ dst,
                          const float* __restrict__ xp, const float* __restrict__ agg,
                          float* __restrict__ outb) {
  int e = blockIdx.x;
  int t = threadIdx.x;           // 0..255 -> (h = t/64, d = t%64)
  int h = t >> 6, d = t & 63;
  int s = src[e], dd = dst[e];
  float alpha = esc[(size_t)e * H_HEADS + h] / (den[dd * H_HEADS + h] + 1e-16f);
  float msg = alpha * (xp[(size_t)s * HD + t] + agg[(size_t)e * 64 + d]);
  atomicAdd(&outb[(size_t)dd * HD + t], msg);
}

__global__ void k_gather_hp(const float* __restrict__ outb, const int* __restrict__ tgt,
                            float* __restrict__ hp) {
  int i = blockIdx.x, t = threadIdx.x;
  hp[(size_t)i * HD + t] = outb[(size_t)tgt[i] * HD + t];
}

__global__ void k_fill(float* __restrict__ p, float val, int n) {
  int i = blockIdx.x * blockDim.x + threadIdx.x;
  if (i < n) p[i] = val;
}

// Semantic attention: e_p = (colsum_p / NT) . wa_p ; beta = softmax(e0,e1). One wave.
__global__ void k_beta(const float* __restrict__ sp, const float* __restrict__ wa0,
                       const float* __restrict__ wa1, float* __restrict__ beta) {
  int l = threadIdx.x;
  float p0 = sp[l] * wa0[l] + sp[l + 32] * wa0[l + 32];
  float p1 = sp[64 + l] * wa1[l] + sp[64 + l + 32] * wa1[l + 32];
  for (int off = 16; off > 0; off >>= 1) {
    p0 += __shfl_xor(p0, off, 32);
    p1 += __shfl_xor(p1, off, 32);
  }
  if (l == 0) {
    float e0 = p0 / (float)NT_TGT, e1 = p1 / (float)NT_TGT;
    float m = fmaxf(e0, e1);
    float x0 = __expf(e0 - m), x1 = __expf(e1 - m);
    float inv = 1.f / (x0 + x1);
    beta[0] = x0 * inv;
    beta[1] = x1 * inv;
  }
}

extern "C" void kernel_launch(void* const* d_in, const int* in_sizes, int n_in,
                              void* d_out, int out_size, void* d_ws, size_t ws_size,
                              hipStream_t stream) {
  const float* x   = (const float*)d_in[0];
  const int*   tgt = (const int*)d_in[5];
  const float* Wo  = (const float*)d_in[20];
  const float* bo  = (const float*)d_in[21];

  float* ws = (float*)d_ws;
  size_t off = 0;
  auto alloc = [&](size_t nf) { float* p = ws + off; off += (nf + 63) & ~(size_t)63; return p; };
  float* xp   = alloc((size_t)N_NODES * HD);     // per-head projections
  float* aggb = alloc((size_t)E_EDGES * D_HID);  // metapath mean features
  float* outb = alloc((size_t)N_NODES * HD);     // attention output accumulator
  float* h0   = alloc((size_t)NT_TGT * HD);
  float* h1   = alloc((size_t)NT_TGT * HD);
  float* ssrc = alloc((size_t)N_NODES * H_HEADS);
  float* sdst = alloc((size_t)N_NODES * H_HEADS);
  float* esc  = alloc((size_t)E_EDGES * H_HEADS);
  float* mmax = alloc((size_t)N_NODES * H_HEADS);
  float* den  = alloc((size_t)N_NODES * H_HEADS);
  float* sp   = alloc(128);                      // [2][64] tanh column sums
  float* beta = alloc(2);

  hipMemsetAsync(sp, 0, 2 * 64 * sizeof(float), stream);

  for (int p = 0; p < 2; ++p) {
    const int* ei  = (const int*)d_in[1 + 2 * p];
    const int* mp  = (const int*)d_in[2 + 2 * p];
    const int* src = ei;
    const int* dst = ei + E_EDGES;
    const float* Wat    = (const float*)d_in[6 + 7 * p + 0];
    const float* a_src  = (const float*)d_in[6 + 7 * p + 1];
    const float* a_dst  = (const float*)d_in[6 + 7 * p + 2];
    const float* a_edge = (const float*)d_in[6 + 7 * p + 3];
    const float* Wsp    = (const float*)d_in[6 + 7 * p + 4];
    const float* bsp    = (const float*)d_in[6 + 7 * p + 5];
    float* hp = (p == 0) ? h0 : h1;

    hipMemsetAsync(outb, 0, (size_t)N_NODES * HD * sizeof(float), stream);
    hipMemsetAsync(den, 0, (size_t)N_NODES * H_HEADS * sizeof(float), stream);
    k_fill<<<(N_NODES * H_HEADS + 255) / 256, 256, 0, stream>>>(
        mmax, -__builtin_huge_valf(), N_NODES * H_HEADS);

    {   // xp = x @ W_attn : [100000,64] x [64,256]
      int tiles = (N_NODES / 16) * (HD / 16);
      k_wmma_gemm<<<(tiles + 3) / 4, 128, 0, stream>>>(x, Wat, xp, N_NODES, D_HID, HD);
    }
    k_node_scores<<<(N_NODES + 7) / 8, 256, 0, stream>>>(xp, a_src, a_dst, ssrc, sdst);
    k_agg_edge<<<(E_EDGES + 7) / 8, 256, 0, stream>>>(x, mp, a_edge, aggb, esc);
    k_edge_score_max<<<(E_EDGES * H_HEADS + 255) / 256, 256, 0, stream>>>(
        esc, src, dst, ssrc, sdst, mmax);
    k_edge_exp<<<(E_EDGES * H_HEADS + 255) / 256, 256, 0, stream>>>(esc, dst, mmax, den);
    k_scatter<<<E_EDGES, 256, 0, stream>>>(esc, den, src, dst, xp, aggb, outb);
    k_gather_hp<<<NT_TGT, 256, 0, stream>>>(outb, tgt, hp);
    {   // column sums of tanh(h_p @ Ws + bs)
      int tiles = (NT_TGT / 16) * (D_HID / 16);
      k_wmma_tanh_reduce<<<(tiles + 3) / 4, 128, 0, stream>>>(hp, Wsp, bsp, sp + p * 64);
    }
  }
  k_beta<<<1, 32, 0, stream>>>(sp, (const float*)d_in[12], (const float*)d_in[19], beta);
  {   // relu((b0*h0+b1*h1) @ Wo + bo)
    int tiles = (NT_TGT / 16) * (D_HID / 16);
    k_wmma_final<<<(tiles + 3) / 4, 128, 0, stream>>>(h0, h1, beta, Wo, bo, (float*)d_out);
  }
}